// Quantizer_module_17635135717565
// MI455X (gfx1250) — compile-verified
//
#include <hip/hip_runtime.h>

// VQ nearest-code search, fused GEMM+argmin for MI455X (gfx1250, wave32).
//   argmin_k ||x-e_k||^2 == argmin_k (||e_k||^2 - 2 x.e_k)
// - v_wmma_f32_16x16x4_f32 over K=512, one 16x16 C tile per wave, 128 tokens/WG.
// - X tile (256KB) + double-buffered 16-code tiles (2x32KB) in LDS, all filled
//   with GLOBAL_LOAD_ASYNC_TO_LDS_B64 (ASYNCcnt) in fragment-major layout:
//     float(row,k) at ((k>>2)*32 + (row&15) + ((k&2)<<3))*2 + (k&1)
//   => per-lane fragment read = base + lane*8 + k0q*256 bytes: conflict-free,
//      uniform stride => ds_load_2addr_b64 pairing, no padding needed.
// - double buffer: s_wait_asynccnt 16 + split barriers overlap DMA with WMMA.

#define N_E   8192
#define E_DIM 512
#define N_TOK 65536
#define MT    128          // tokens per workgroup (8 waves x 16 rows)
#define NT    16           // codes per N-iteration
#define NTILE (N_E / NT)   // 512

typedef float v2f __attribute__((ext_vector_type(2)));
typedef float v8f __attribute__((ext_vector_type(8)));

__global__ __launch_bounds__(32) void en2_kernel(const float* __restrict__ emb,
                                                 float* __restrict__ en2) {
  const int code = blockIdx.x;
  const int lane = threadIdx.x;
  const float4* e4 = (const float4*)(emb + (size_t)code * E_DIM);
  float s = 0.f;
  #pragma unroll
  for (int i = lane; i < E_DIM / 4; i += 32) {
    float4 v = e4[i];
    s += v.x * v.x + v.y * v.y + v.z * v.z + v.w * v.w;
  }
  #pragma unroll
  for (int m = 16; m >= 1; m >>= 1) s += __shfl_xor(s, m, 32);
  if (lane == 0) en2[code] = s;
}

// one async B64 transfer: 8B/lane, LDS dest addr + 32-bit global offset + SGPR base
__device__ __forceinline__ void async_b64(unsigned lds_addr, unsigned goff,
                                          const void* sbase) {
  asm volatile("global_load_async_to_lds_b64 %0, %1, %2"
               :: "v"(lds_addr), "v"(goff), "s"(sbase) : "memory");
}

// issue one 16-row x 512-float tile (32KB) in fragment-major layout; 16 instrs/wave
__device__ __forceinline__ void issue_tile16(unsigned lds_base, int p,
                                             const void* gsrc) {
  const unsigned vdst = lds_base + ((unsigned)(p >> 1) << 8) + ((unsigned)(p & 1) << 7);
  const unsigned voff = (unsigned)p << 3;
  #pragma unroll
  for (int j = 0; j < 16; ++j)                   // j = row within tile
    async_b64(vdst + j * 8u, voff + j * 2048u, gsrc);
}

__global__ __launch_bounds__(256) void vq_argmin_kernel(
    const float* __restrict__ x, const float* __restrict__ emb,
    const float* __restrict__ en2g, float* __restrict__ zq,
    int* __restrict__ out_idx) {
  extern __shared__ float smem[];                // Xf: 65536 f, EsA/EsB: 8192 f each
  const int tid  = threadIdx.x;
  const int lane = tid & 31;
  const int wave = tid >> 5;
  const int tok0 = blockIdx.x * MT;
  const int p    = wave * 32 + lane;             // this thread's 8B column slice

  const unsigned ldsX  = (unsigned)(uintptr_t)(void*)smem;
  const unsigned ldsEA = ldsX + 262144u;
  const unsigned ldsEB = ldsX + 262144u + 32768u;

  // ---- async fill of X tile (128 rows, fragment-major), coalesced 256B/instr ----
  {
    const void* gx = (const void*)(x + (size_t)tok0 * E_DIM);
    const unsigned vdst = ldsX + ((unsigned)(p >> 1) << 8) + ((unsigned)(p & 1) << 7);
    const unsigned voff = (unsigned)p << 3;
    #pragma unroll
    for (int j = 0; j < MT; ++j)                 // j = token row
      async_b64(vdst + (unsigned)(j >> 4) * 32768u + (unsigned)(j & 15) * 8u,
                voff + (unsigned)j * 2048u, gx);
  }
  // ---- prefetch code tiles 0 and 1 ----
  issue_tile16(ldsEA, p, (const void*)emb);
  issue_tile16(ldsEB, p, (const void*)(emb + (size_t)NT * E_DIM));

  float minv[8];
  int   mini[8];
  #pragma unroll
  for (int j = 0; j < 8; ++j) { minv[j] = 3.4e38f; mini[j] = 0; }

  const float* pa = smem + wave * 8192 + lane * 2;  // this wave's A fragments

  for (int nt = 0; nt < NTILE; ++nt) {
    const int nbase = nt * NT;
    const float e2 = en2g[nbase + (lane & 15)];
    const float* pb = smem + 65536 + (nt & 1) * 8192 + lane * 2;

    asm volatile("s_wait_asynccnt 0x10" ::: "memory");  // tile nt landed (next 16 in flight)
    __syncthreads();

    v8f acc = {};
    #pragma unroll 8
    for (int k0q = 0; k0q < E_DIM / 4; ++k0q) {
      v2f a = *(const v2f*)(pa + (k0q << 6));
      v2f b = *(const v2f*)(pb + (k0q << 6));
      acc = __builtin_amdgcn_wmma_f32_16x16x4_f32(
          false, a, false, b, (short)0, acc, false, false);
    }

    __syncthreads();                             // all waves done reading buf[nt&1]
    {                                            // refill it with tile nt+2 (wraps)
      const int n2 = (nt + 2) & (NTILE - 1);
      issue_tile16((nt & 1) ? ldsEB : ldsEA, p,
                   (const void*)(emb + (size_t)n2 * NT * E_DIM));
    }

    const int ncol = nbase + (lane & 15);
    #pragma unroll
    for (int j = 0; j < 8; ++j) {
      float sc = fmaf(-2.0f, acc[j], e2);
      if (sc < minv[j]) { minv[j] = sc; mini[j] = ncol; }  // strict <: first idx wins
    }
  }

  // ---- argmin across the 16 lanes holding one row (wave32 butterfly) ----
  int redid[8];
  #pragma unroll
  for (int j = 0; j < 8; ++j) {
    float v = minv[j];
    int   id = mini[j];
    #pragma unroll
    for (int m = 1; m <= 8; m <<= 1) {
      float ov = __shfl_xor(v, m, 32);
      int   oi = __shfl_xor(id, m, 32);
      if (ov < v || (ov == v && oi < id)) { v = ov; id = oi; }
    }
    redid[j] = id;                               // valid in every lane of each half
  }

  // ---- write indices + gather z_q = emb[idx] (emb rows L2-resident) ----
  #pragma unroll
  for (int r = 0; r < 16; ++r) {
    const int idr = __shfl(redid[r & 7], (r >> 3) << 4, 32);  // lane 0 / lane 16
    const int row = tok0 + wave * 16 + r;
    if (lane == 0) out_idx[row] = idr;
    const float4* er = (const float4*)(emb + (size_t)idr * E_DIM);
    float4* zr = (float4*)(zq + (size_t)row * E_DIM);
    for (int c = lane; c < E_DIM / 4; c += 32) zr[c] = er[c];
  }
}

extern "C" void kernel_launch(void* const* d_in, const int* in_sizes, int n_in,
                              void* d_out, int out_size, void* d_ws, size_t ws_size,
                              hipStream_t stream) {
  (void)in_sizes; (void)n_in; (void)out_size; (void)ws_size;
  const float* x   = (const float*)d_in[0];   // [N_TOK, E_DIM] fp32
  const float* emb = (const float*)d_in[1];   // [N_E, E_DIM] fp32
  float* en2 = (float*)d_ws;                  // 8192 floats scratch
  float* zq  = (float*)d_out;                 // [N_TOK, E_DIM] fp32
  int* out_idx = (int*)((float*)d_out + (size_t)N_TOK * E_DIM);  // [N_TOK] int32

  en2_kernel<<<N_E, 32, 0, stream>>>(emb, en2);

  size_t shbytes = 262144u + 2u * 32768u;     // 320 KB exactly (CDNA5 WG max)
  vq_argmin_kernel<<<N_TOK / MT, 256, shbytes, stream>>>(x, emb, en2, zq, out_idx);
}